// LSHAttention_56873956933958
// MI455X (gfx1250) — compile-verified
//
#include <hip/hip_runtime.h>
#include <hip/hip_bf16.h>

// LSH attention bucketing for MI455X (gfx1250, wave32).
// Phase 1: per-batch projection GEMM (4096x1024)x(1024x32) via V_WMMA_F32_16X16X4_F32,
//          then per-token argmax over [x, -x] -> bucket id.
// Phase 2: stable counting sort of token indices by bucket id (== jnp.argsort).

typedef __attribute__((ext_vector_type(2))) float v2f;
typedef __attribute__((ext_vector_type(8))) float v8f;

#define BATCH 4
#define SEQ   4096
#define DIM   1024
#define NPROJ 32          // BUCKET_SIZE / 2
#define NBUCKET 64        // BUCKET_SIZE

// ---------------------------------------------------------------------------
// Kernel 1: one wave (32 lanes) computes a 16(token) x 32(proj) tile with WMMA,
// then the bucket id for each of its 16 tokens.
// Grid: (SEQ/16, BATCH), Block: 32 threads (exactly one wave; EXEC all-1s for WMMA).
// ---------------------------------------------------------------------------
__global__ __launch_bounds__(32)
void lsh_project_bucket_kernel(const float* __restrict__ Q,
                               const float* __restrict__ R,
                               int* __restrict__ buckets) {
    __shared__ float proj[16][NPROJ + 1];   // +1 pad: conflict-free row reads

    const int lane    = threadIdx.x;        // 0..31
    const int m0      = blockIdx.x * 16;    // token tile base
    const int b       = blockIdx.y;         // batch
    const int halfSel = lane >> 4;          // 0 -> K={0,1}, 1 -> K={2,3}
    const int mrow    = lane & 15;          // A-matrix row handled by this lane
    const int n       = lane & 15;          // B/C-matrix column handled by this lane
    const int klo     = halfSel * 2;

    // A: Q[b, m0+mrow, klo + k .. klo + k + 1]  (ISA 32-bit 16x4 A layout)
    const float* qrow = Q + ((size_t)b * SEQ + (size_t)(m0 + mrow)) * DIM + klo;
    // B: R[b, klo + k, n] / R[b, klo + k + 1, n]  (4x16 B layout, two N-tiles)
    const float* rcol0 = R + ((size_t)b * DIM + (size_t)klo) * NPROJ + n;   // cols  0..15
    const float* rcol1 = rcol0 + 16;                                         // cols 16..31

    v8f acc0 = {};   // C/D for columns  0..15
    v8f acc1 = {};   // C/D for columns 16..31

    for (int k = 0; k < DIM; k += 4) {
        v2f a = *(const v2f*)(qrow + k);

        v2f b0, b1;
        b0.x = rcol0[(size_t)(k + 0) * NPROJ];
        b0.y = rcol0[(size_t)(k + 1) * NPROJ];
        b1.x = rcol1[(size_t)(k + 0) * NPROJ];
        b1.y = rcol1[(size_t)(k + 1) * NPROJ];

        // 8-arg form: (neg_a, A, neg_b, B, c_mod, C, reuse_a, reuse_b)
        acc0 = __builtin_amdgcn_wmma_f32_16x16x4_f32(false, a, false, b0,
                                                     (short)0, acc0, false, false);
        acc1 = __builtin_amdgcn_wmma_f32_16x16x4_f32(false, a, false, b1,
                                                     (short)0, acc1, false, false);
    }

    // Spill accumulators to LDS per the documented C layout:
    // VGPR i, lanes 0-15 -> row i,   col = lane
    // VGPR i, lanes 16-31 -> row i+8, col = lane-16
    const int rbase = halfSel * 8;
#pragma unroll
    for (int i = 0; i < 8; ++i) {
        proj[rbase + i][n]      = acc0[i];
        proj[rbase + i][n + 16] = acc1[i];
    }
    __syncthreads();

    // 16 lanes: bucket = argmax over concat([x, -x]) of length 64.
    // First-occurrence tie-breaking matches jnp.argmax: jmax always < 32+jmin,
    // so ">=" on the max-vs-(-min) comparison is exact.
    if (lane < 16) {
        float maxv = proj[lane][0]; int maxi = 0;
        float minv = maxv;          int mini = 0;
#pragma unroll
        for (int j = 1; j < NPROJ; ++j) {
            float v = proj[lane][j];
            if (v > maxv) { maxv = v; maxi = j; }
            if (v < minv) { minv = v; mini = j; }
        }
        const int bucket = (maxv >= -minv) ? maxi : (NPROJ + mini);
        buckets[(size_t)b * SEQ + (size_t)(m0 + lane)] = bucket;
    }
}

// ---------------------------------------------------------------------------
// Kernel 2: stable counting sort per batch row -> sticker = argsort(buckets).
// Grid: BATCH blocks, 64 threads (one thread per bucket value).
// ---------------------------------------------------------------------------
__global__ __launch_bounds__(64)
void lsh_bucket_sort_kernel(const int* __restrict__ buckets,
                            float* __restrict__ sticker) {
    __shared__ int lb[SEQ];
    __shared__ int counts[NBUCKET];

    const int t = threadIdx.x;   // bucket value owned by this thread
    const int b = blockIdx.x;

    for (int s = t; s < SEQ; s += NBUCKET)
        lb[s] = buckets[(size_t)b * SEQ + s];
    __syncthreads();

    int cnt = 0;
    for (int s = 0; s < SEQ; ++s)
        cnt += (lb[s] == t) ? 1 : 0;
    counts[t] = cnt;
    __syncthreads();

    int off = 0;                 // exclusive prefix sum over smaller buckets
    for (int i = 0; i < t; ++i) off += counts[i];

    // Scatter token indices of this bucket in ascending order (stable).
    for (int s = 0; s < SEQ; ++s) {
        if (lb[s] == t) {
            sticker[(size_t)b * SEQ + off] = (float)s;
            ++off;
        }
    }
}

extern "C" void kernel_launch(void* const* d_in, const int* in_sizes, int n_in,
                              void* d_out, int out_size, void* d_ws, size_t ws_size,
                              hipStream_t stream) {
    const float* Q = (const float*)d_in[0];   // query [B,S,D]
    // d_in[1] = key, d_in[2] = value: dead in the reference (only sticker returned)
    const float* R = (const float*)d_in[3];   // hyperplanes [B,D,32]

    int*   buckets = (int*)d_ws;              // [B,S] = 64 KB scratch
    float* out     = (float*)d_out;           // [B,S] sticker indices

    dim3 grid1(SEQ / 16, BATCH);
    lsh_project_bucket_kernel<<<grid1, 32, 0, stream>>>(Q, R, buckets);
    lsh_bucket_sort_kernel<<<BATCH, NBUCKET, 0, stream>>>(buckets, out);
}